// ISKGModel_85882166051091
// MI455X (gfx1250) — compile-verified
//
#include <hip/hip_runtime.h>
#include <hip/hip_bf16.h>

#define N_NODES  150000
#define N_USER   100000
#define DIM      64
#define NLAYERS  3
#define BATCH    4096
#define NEG_SLOPE 0.01f

typedef float v2f __attribute__((ext_vector_type(2)));
typedef float v8f __attribute__((ext_vector_type(8)));

// ---------------------------------------------------------------------------
// SpMM scatter: agg[r] += val * x[c], 16 lanes per edge, float4 per lane.
// Feature matrix (38.4MB) lives in L2 (192MB), so gathers+atomics are L2-local.
// ---------------------------------------------------------------------------
__global__ __launch_bounds__(256) void spmm_scatter_kernel(
    const int* __restrict__ rows, const int* __restrict__ cols,
    const float* __restrict__ vals, const float* __restrict__ x,
    float* __restrict__ agg, int nnz)
{
    long gid = (long)blockIdx.x * blockDim.x + threadIdx.x;
    int e = (int)(gid >> 4);
    int j = (int)(gid & 15);
    if (e >= nnz) return;
    int   r = rows[e];
    int   c = cols[e];
    float v = vals[e];
    const float4* src = (const float4*)(x + (long)c * DIM);
    __builtin_prefetch(src, 0, 0);            // global_prefetch_b8
    float4 mv = src[j];
    float* dst = agg + (long)r * DIM + j * 4;
    atomicAdd(dst + 0, v * mv.x);
    atomicAdd(dst + 1, v * mv.y);
    atomicAdd(dst + 2, v * mv.z);
    atomicAdd(dst + 3, v * mv.w);
}

// ---------------------------------------------------------------------------
// Pack Wt [256,64] into K-pairs: wtp[j*64+col] = {Wt[2j][col], Wt[2j+1][col]}
// so score-kernel B operands are single contiguous b64 loads.
// ---------------------------------------------------------------------------
__global__ __launch_bounds__(256) void pack_wt_kernel(
    const float* __restrict__ Wt, float2* __restrict__ wtp)
{
    int gid = blockIdx.x * 256 + threadIdx.x;   // 0 .. 128*64-1
    int j = gid >> 6;
    int col = gid & 63;
    wtp[gid] = make_float2(Wt[(2 * j) * DIM + col], Wt[(2 * j + 1) * DIM + col]);
}

// ---------------------------------------------------------------------------
// Fused layer: out = leaky_relu((agg+f)@W1 + b1 + (agg*f)@W2 + b2)
// One wave per 16-row tile; V_WMMA_F32_16X16X4_F32, K=64 in 16 steps.
// A layout (16x4 f32): lane<16 -> {A[m][k0], A[m][k0+1]},
//                      lane>=16 -> {A[m][k0+2], A[m][k0+3]}
// B layout (4x16 f32): symmetric; C/D: VGPR r, lane n -> D[r+8*(n/16)][n%16].
// Weights staged in LDS as pre-paired float2 so B operand = one ds_load_b64.
// ---------------------------------------------------------------------------
__global__ __launch_bounds__(256) void gcn_layer_kernel(
    const float* __restrict__ feats, const float* __restrict__ agg,
    const float* __restrict__ W1, const float* __restrict__ b1,
    const float* __restrict__ W2, const float* __restrict__ b2,
    float* __restrict__ outf, int nrows)
{
    __shared__ float2 lwp1[32 * DIM];   // pair j covers rows 2j, 2j+1
    __shared__ float2 lwp2[32 * DIM];
    __shared__ float  lb[DIM];

    int t = threadIdx.x;
    for (int e = t; e < 32 * DIM; e += 256) {
        int j = e >> 6, col = e & 63;
        lwp1[e] = make_float2(W1[(2 * j) * DIM + col], W1[(2 * j + 1) * DIM + col]);
        lwp2[e] = make_float2(W2[(2 * j) * DIM + col], W2[(2 * j + 1) * DIM + col]);
    }
    if (t < DIM) lb[t] = b1[t] + b2[t];
    __syncthreads();

    int wave = t >> 5;
    int lane = t & 31;
    int m = lane & 15;           // row within tile (A) / col within tile (B,D)
    int h = lane >> 4;           // half-wave selector
    int tile = blockIdx.x * 8 + wave;
    int row0 = tile * 16;
    if (row0 >= nrows) return;   // wave-uniform: EXEC stays all-ones for WMMA

    // Preload A tiles: s = agg + feats, p = agg * feats  (32 floats per lane each)
    v2f sA[16], pA[16];
    const float2* frv = (const float2*)(feats + (long)(row0 + m) * DIM);
    const float2* arv = (const float2*)(agg   + (long)(row0 + m) * DIM);
#pragma unroll
    for (int ks = 0; ks < 16; ++ks) {
        int j = ks * 2 + h;              // pair index: rows 2j, 2j+1 of K
        float2 f = frv[j];
        float2 a = arv[j];
        v2f s; s.x = a.x + f.x; s.y = a.y + f.y;
        v2f p; p.x = a.x * f.x; p.y = a.y * f.y;
        sA[ks] = s; pA[ks] = p;
    }

#pragma unroll
    for (int ct = 0; ct < 4; ++ct) {
        int col = ct * 16 + m;
        float bias = lb[col];
        v8f acc;
#pragma unroll
        for (int r = 0; r < 8; ++r) acc[r] = bias;
#pragma unroll
        for (int ks = 0; ks < 16; ++ks) {
            int j = ks * 2 + h;
            float2 w1 = lwp1[j * DIM + col];
            float2 w2 = lwp2[j * DIM + col];
            v2f bv1; bv1.x = w1.x; bv1.y = w1.y;
            v2f bv2; bv2.x = w2.x; bv2.y = w2.y;
            acc = __builtin_amdgcn_wmma_f32_16x16x4_f32(
                false, sA[ks], false, bv1, (short)0, acc, false, false);
            acc = __builtin_amdgcn_wmma_f32_16x16x4_f32(
                false, pA[ks], false, bv2, (short)0, acc, false, false);
        }
#pragma unroll
        for (int r = 0; r < 8; ++r) {
            float v = acc[r];
            v = v > 0.0f ? v : NEG_SLOPE * v;
            int row = row0 + r + 8 * h;
            outf[(long)row * DIM + col] = v;
        }
    }
}

// ---------------------------------------------------------------------------
// Scoring: out[b] = dot(xu@Wt + bt, xi@Wt + bt), K = 256 over 4 emb buffers.
// One wave per 16 batch rows; WMMA f32, then 16-lane shuffle reduction.
// B operands come from pre-paired wtp (contiguous b64, L2 resident).
// ---------------------------------------------------------------------------
__global__ __launch_bounds__(256) void score_kernel(
    const float* __restrict__ e0, const float* __restrict__ e1,
    const float* __restrict__ e2, const float* __restrict__ e3,
    const float2* __restrict__ wtp, const float* __restrict__ bt,
    const int* __restrict__ users, const int* __restrict__ items,
    float* __restrict__ out)
{
    int t = threadIdx.x;
    int wave = t >> 5;
    int lane = t & 31;
    int m = lane & 15;
    int h = lane >> 4;
    int tile = blockIdx.x * 8 + wave;         // 0..255 exactly (32 blocks * 8)
    int b = tile * 16 + m;

    long ru = (long)users[b];                 // user row in emb tables
    long ri = (long)N_USER + (long)items[b];  // item row

    const float* bufs[4] = {e0, e1, e2, e3};

    v8f accU[4], accI[4];
#pragma unroll
    for (int ct = 0; ct < 4; ++ct) {
        float bias = bt[ct * 16 + m];
#pragma unroll
        for (int r = 0; r < 8; ++r) { accU[ct][r] = bias; accI[ct][r] = bias; }
    }

#pragma unroll
    for (int q = 0; q < 4; ++q) {             // which embedding buffer (K/64)
        const float2* rU = (const float2*)(bufs[q] + ru * DIM);
        const float2* rI = (const float2*)(bufs[q] + ri * DIM);
#pragma unroll
        for (int ks = 0; ks < 16; ++ks) {
            int j  = ks * 2 + h;              // pair within buffer
            int jg = q * 32 + j;              // global K-pair index for wtp
            float2 fu = rU[j];
            float2 fi = rI[j];
            v2f aU; aU.x = fu.x; aU.y = fu.y;
            v2f aI; aI.x = fi.x; aI.y = fi.y;
#pragma unroll
            for (int ct = 0; ct < 4; ++ct) {
                float2 w = wtp[jg * DIM + ct * 16 + m];
                v2f bv; bv.x = w.x; bv.y = w.y;
                accU[ct] = __builtin_amdgcn_wmma_f32_16x16x4_f32(
                    false, aU, false, bv, (short)0, accU[ct], false, false);
                accI[ct] = __builtin_amdgcn_wmma_f32_16x16x4_f32(
                    false, aI, false, bv, (short)0, accI[ct], false, false);
            }
        }
    }

    // out[row] = sum_n u[n]*i[n]; row = tile*16 + r + 8*h, cols across 16 lanes
    float pr[8];
#pragma unroll
    for (int r = 0; r < 8; ++r) {
        float s = 0.0f;
#pragma unroll
        for (int ct = 0; ct < 4; ++ct) s += accU[ct][r] * accI[ct][r];
        pr[r] = s;
    }
#pragma unroll
    for (int mask = 1; mask < 16; mask <<= 1) {
#pragma unroll
        for (int r = 0; r < 8; ++r) pr[r] += __shfl_xor(pr[r], mask, 32);
    }
    if (m == 0) {
#pragma unroll
        for (int r = 0; r < 8; ++r) out[tile * 16 + 8 * h + r] = pr[r];
    }
}

// ---------------------------------------------------------------------------
extern "C" void kernel_launch(void* const* d_in, const int* in_sizes, int n_in,
                              void* d_out, int out_size, void* d_ws, size_t ws_size,
                              hipStream_t stream)
{
    const float* features  = (const float*)d_in[0];
    const int*   edge_rows = (const int*)  d_in[1];
    const int*   edge_cols = (const int*)  d_in[2];
    const float* edge_vals = (const float*)d_in[3];
    const float* W1s       = (const float*)d_in[4];
    const float* b1s       = (const float*)d_in[5];
    const float* W2s       = (const float*)d_in[6];
    const float* b2s       = (const float*)d_in[7];
    const float* Wt        = (const float*)d_in[8];
    const float* bt        = (const float*)d_in[9];
    const int*   users     = (const int*)  d_in[10];
    const int*   items     = (const int*)  d_in[11];
    float* out = (float*)d_out;

    int nnz = in_sizes[1];

    size_t nfeat = (size_t)N_NODES * DIM;
    float* agg = (float*)d_ws;
    float* e1  = agg + nfeat;
    float* e2  = e1 + nfeat;
    float* e3  = e2 + nfeat;
    float2* wtp = (float2*)(e3 + nfeat);        // 128*64 float2 = 64KB
    float* layer_out[NLAYERS] = {e1, e2, e3};
    const float* layer_in[NLAYERS] = {features, e1, e2};

    dim3 spmm_grid((unsigned)(((long)nnz * 16 + 255) / 256));
    int tiles = (N_NODES + 15) / 16;            // 9375
    int gcn_blocks = (tiles + 7) / 8;           // 1172

    pack_wt_kernel<<<(128 * DIM) / 256, 256, 0, stream>>>(Wt, wtp);

    for (int l = 0; l < NLAYERS; ++l) {
        hipMemsetAsync(agg, 0, nfeat * sizeof(float), stream);
        spmm_scatter_kernel<<<spmm_grid, 256, 0, stream>>>(
            edge_rows, edge_cols, edge_vals, layer_in[l], agg, nnz);
        gcn_layer_kernel<<<gcn_blocks, 256, 0, stream>>>(
            layer_in[l], agg,
            W1s + (size_t)l * DIM * DIM, b1s + (size_t)l * DIM,
            W2s + (size_t)l * DIM * DIM, b2s + (size_t)l * DIM,
            layer_out[l], N_NODES);
    }

    score_kernel<<<(BATCH / 16) / 8, 256, 0, stream>>>(
        features, e1, e2, e3, wtp, bt, users, items, out);
}